// TrainableFilter_75118978007282
// MI455X (gfx1250) — compile-verified
//
#include <hip/hip_runtime.h>
#include <cstdint>

// Bilateral 5x5 trainable filter, x:[4,3,512,512] f32, ws:[5,5] f32 -> out f32.
// Density normalization cancels algebraically: out = sum(ws*e*p)/sum(ws*e).
// Packed-fp32 (VOP3P v_pk_*) over column pairs; hardware exp2; async-to-LDS staging.

typedef float v2f __attribute__((ext_vector_type(2)));

constexpr int IMG_H = 512;
constexpr int IMG_W = 512;
constexpr int TW    = 64;            // output tile width
constexpr int TH    = 16;            // output tile height
constexpr int PADR  = 2;             // (5-1)/2
constexpr int SW    = TW + 2*PADR;   // 68
constexpr int SH    = TH + 2*PADR;   // 20
constexpr int NTHR  = 256;           // 8 waves (wave32)
constexpr int NSTAGE = (SH*SW + NTHR - 1) / NTHR; // 6

__device__ __forceinline__ int reflect512(int t) {
  t = (t < 0) ? -t : t;                      // np.pad 'reflect': -1->1, -2->2
  t = (t >= IMG_H) ? (2*IMG_H - 2 - t) : t;  // 512->510, 513->509
  return t;
}

__global__ __launch_bounds__(NTHR)
void bilateral5x5_kernel(const float* __restrict__ x,
                         const float* __restrict__ wspace,
                         float* __restrict__ out) {
  __shared__ float tile[SH][SW];

  const int ch  = blockIdx.z;
  const int tx0 = blockIdx.x * TW;
  const int ty0 = blockIdx.y * TH;
  const int tid = threadIdx.x;

  const float* __restrict__ xc = x + (size_t)ch * (IMG_H * IMG_W);

  // ---- Stage halo tile into LDS with gfx1250 async copies (ASYNCcnt). ----
  // Fully unrolled: 6 static back-to-back issues per lane; reflect boundary
  // mapping folds into each lane's global address; per-lane LDS destination.
  #pragma unroll
  for (int k = 0; k < NSTAGE; ++k) {
    const int i = tid + k * NTHR;
    if (i < SH * SW) {
      const int ly = i / SW;
      const int lx = i - ly * SW;
      const int gy = reflect512(ty0 + ly - PADR);
      const int gx = reflect512(tx0 + lx - PADR);
      const float*   gp = xc + gy * IMG_W + gx;
      const uint32_t lp = (uint32_t)(uintptr_t)(&tile[ly][lx]); // LDS byte addr
      asm volatile("global_load_async_to_lds_b32 %0, %1, off"
                   :: "v"(lp), "v"(gp) : "memory");
    }
  }
  asm volatile("s_wait_asynccnt 0" ::: "memory");
  __syncthreads();

  // Spatial weights: uniform pointer + constant indices -> scalar loads.
  // Pairs match the packed column layout.
  v2f ws01[5], ws23[5]; float ws4[5];
  #pragma unroll
  for (int ki = 0; ki < 5; ++ki) {
    ws01[ki].x = wspace[5*ki + 0]; ws01[ki].y = wspace[5*ki + 1];
    ws23[ki].x = wspace[5*ki + 2]; ws23[ki].y = wspace[5*ki + 3];
    ws4[ki]    = wspace[5*ki + 4];
  }

  const int lx  = tid & (TW - 1);   // 0..63 (coalesced along rows)
  const int ly0 = (tid >> 6) * 4;   // 4 consecutive output rows per thread

  // Register window: 8 rows x 5 cols, columns held as {pair01, pair23, col4}.
  // ds_load_b32 writes straight into the 64-bit vreg halves: packing is free.
  v2f w01[8], w23[8]; float w4[8];
  #pragma unroll
  for (int a = 0; a < 8; ++a) {
    w01[a].x = tile[ly0 + a][lx + 0];
    w01[a].y = tile[ly0 + a][lx + 1];
    w23[a].x = tile[ly0 + a][lx + 2];
    w23[a].y = tile[ly0 + a][lx + 3];
    w4[a]    = tile[ly0 + a][lx + 4];
  }

  // exp(-d^2/2) = exp2(d^2 * -log2(e)/2) -> pk_mul, pk_mul, then v_exp_f32/half
  constexpr float KEXP = -0.72134752044448170f;
  const v2f K2 = {KEXP, KEXP};

  #pragma unroll
  for (int r = 0; r < 4; ++r) {
    const float c  = w23[r + 2].x;         // center pixel (col 2 of row r+2)
    const v2f   cc = {c, c};
    v2f   den2 = {0.f, 0.f}, num2 = {0.f, 0.f};
    float den1 = 0.f,        num1 = 0.f;

    #pragma unroll
    for (int ki = 0; ki < 5; ++ki) {
      const int a = r + ki;
      { // columns 0,1 (packed)
        const v2f p = w01[a];
        const v2f d = p - cc;
        const v2f t = d * d * K2;
        v2f e; e.x = __builtin_amdgcn_exp2f(t.x);
               e.y = __builtin_amdgcn_exp2f(t.y);
        den2 = __builtin_elementwise_fma(e,     ws01[ki], den2);
        num2 = __builtin_elementwise_fma(e * p, ws01[ki], num2);
      }
      { // columns 2,3 (packed)
        const v2f p = w23[a];
        const v2f d = p - cc;
        const v2f t = d * d * K2;
        v2f e; e.x = __builtin_amdgcn_exp2f(t.x);
               e.y = __builtin_amdgcn_exp2f(t.y);
        den2 = __builtin_elementwise_fma(e,     ws23[ki], den2);
        num2 = __builtin_elementwise_fma(e * p, ws23[ki], num2);
      }
      { // column 4 (scalar)
        const float p = w4[a];
        const float d = p - c;
        const float e = __builtin_amdgcn_exp2f(d * d * KEXP);
        den1 = fmaf(e,     ws4[ki], den1);
        num1 = fmaf(e * p, ws4[ki], num1);
      }
    }

    const float den = den2.x + den2.y + den1;
    const float num = num2.x + num2.y + num1;
    out[((size_t)ch * IMG_H + (ty0 + ly0 + r)) * IMG_W + (tx0 + lx)] = num / den;
  }
}

extern "C" void kernel_launch(void* const* d_in, const int* in_sizes, int n_in,
                              void* d_out, int out_size, void* d_ws, size_t ws_size,
                              hipStream_t stream) {
  const float* x  = (const float*)d_in[0];
  const float* ws = (const float*)d_in[1];
  float* out      = (float*)d_out;

  const int chans = out_size / (IMG_H * IMG_W);   // B*C = 12
  dim3 grid(IMG_W / TW, IMG_H / TH, chans);       // 8 x 32 x 12
  dim3 block(NTHR);
  hipLaunchKernelGGL(bilateral5x5_kernel, grid, block, 0, stream, x, ws, out);
}